// center_loss_8744553414792
// MI455X (gfx1250) — compile-verified
//
#include <hip/hip_runtime.h>
#include <math.h>

typedef float v2f __attribute__((ext_vector_type(2)));
typedef float v4f __attribute__((ext_vector_type(4)));
typedef float v8f __attribute__((ext_vector_type(8)));

#define N_ROWS          65536
#define DDIM            512
#define ROWS_PER_WAVE   16
#define WAVES_PER_BLOCK 8
#define CHUNK           64
#define LDS_STRIDE      68   // 64 + 4 floats pad -> conflict-free strided A reads

__launch_bounds__(256)
__global__ void center_loss_kernel(const float* __restrict__ feature,
                                   const int*   __restrict__ label,
                                   const float* __restrict__ centers,
                                   float*       __restrict__ out) {
    __shared__ __align__(16) float tile[WAVES_PER_BLOCK * ROWS_PER_WAVE * LDS_STRIDE];
    __shared__ float blockAcc;

    const int tid  = threadIdx.x;
    const int wave = tid >> 5;
    const int lane = tid & 31;
    const int half = lane >> 4;     // 0 or 1 (A-matrix K half per ISA layout)
    const int m    = lane & 15;     // WMMA A row; also column-group for staging
    const int rowBase = (blockIdx.x * WAVES_PER_BLOCK + wave) * ROWS_PER_WAVE;

    if (tid == 0) blockAcc = 0.0f;

    float* wtile = &tile[wave * ROWS_PER_WAVE * LDS_STRIDE];

    // Staging iteration i handles row r = half + 2*i, float columns [4m, 4m+4).
    // Lanes 0-15 cover one 256B row segment, lanes 16-31 the next row: coalesced.
    const float* fptr[8];
    const float* cptr[8];
    int          rloc[8];
#pragma unroll
    for (int i = 0; i < 8; ++i) {
        const int r = half + 2 * i;
        rloc[i] = r;
        const int g = rowBase + r;
        fptr[i] = feature + (size_t)g * DDIM + m * 4;
        const int lbl = label[g];                 // small, L2-resident
        cptr[i] = centers + (size_t)lbl * DDIM + m * 4;
    }

    v2f onesB;  onesB[0] = 1.0f; onesB[1] = 1.0f;   // B = 4x16 all-ones
    v8f acc = {0.f, 0.f, 0.f, 0.f, 0.f, 0.f, 0.f, 0.f};

    for (int k0 = 0; k0 < DDIM; k0 += CHUNK) {
        // ---- Stage diff^2 tile (16 rows x 64 cols), fully coalesced ----
#pragma unroll
        for (int i = 0; i < 8; ++i) {
            // feature: read-once 128MB stream -> non-temporal (keep centers hot in L2)
            v4f f = __builtin_nontemporal_load((const v4f*)(fptr[i] + k0));
            v4f c = *(const v4f*)(cptr[i] + k0);
            v4f d = f - c;
            d = d * d;
            *(v4f*)(wtile + rloc[i] * LDS_STRIDE + m * 4) = d;
        }
        __syncthreads();

        // ---- Reduce 16x64 tile with 16 x V_WMMA_F32_16X16X4_F32 (B = ones) ----
        // A layout (ISA 7.12.2): lane = half*16 + m holds A[m][2*half], A[m][2*half+1]
#pragma unroll
        for (int kk = 0; kk < 16; ++kk) {
            v2f a = *(const v2f*)(wtile + m * LDS_STRIDE + kk * 4 + 2 * half);
            acc = __builtin_amdgcn_wmma_f32_16x16x4_f32(
                      false, a, false, onesB, (short)0, acc, false, false);
        }
        __syncthreads();
    }

    // C/D layout: VGPR v -> lanes 0-15: M=v (rows 0-7), lanes 16-31: M=v+8 (rows 8-15).
    // Every N column is identical (B all-ones), so lanes 0 and 16 suffice.
    if (m == 0) {
        float s = 0.0f;
#pragma unroll
        for (int v = 0; v < 8; ++v) s += sqrtf(acc[v]);
        atomicAdd(&blockAcc, s);                  // ds_add_f32, 16 per block
    }
    __syncthreads();
    if (tid == 0)
        atomicAdd(out, blockAcc * (1.0f / (float)N_ROWS));  // 512 global atomics total
}

extern "C" void kernel_launch(void* const* d_in, const int* in_sizes, int n_in,
                              void* d_out, int out_size, void* d_ws, size_t ws_size,
                              hipStream_t stream) {
    const float* feature = (const float*)d_in[0];
    const int*   label   = (const int*)d_in[1];
    const float* centers = (const float*)d_in[2];
    float*       out     = (float*)d_out;

    hipMemsetAsync(d_out, 0, sizeof(float), stream);  // graph-capturable

    const int rows_per_block = ROWS_PER_WAVE * WAVES_PER_BLOCK;   // 128
    dim3 grid(N_ROWS / rows_per_block);                           // 512 blocks
    dim3 block(256);
    center_loss_kernel<<<grid, block, 0, stream>>>(feature, label, centers, out);
}